// QCDAwareNoNorm_68813966017011
// MI455X (gfx1250) — compile-verified
//
#include <hip/hip_runtime.h>
#include <hip/hip_bf16.h>

typedef __attribute__((ext_vector_type(2))) float v2f;
typedef __attribute__((ext_vector_type(8))) float v8f;

#define NCOL 2048
#define ROWS 16
#define EPSF 1e-10f

// One block handles (batch b, 16-row strip i0..i0+15) of the 2048x2048 softmax
// matrix. 256 threads = 8 waves; each wave owns 16 column tiles of 16.
__global__ __launch_bounds__(256) void qcd_softmax_kernel(
    const float* __restrict__ emb,   // (8, 6, 2048)
    const float* __restrict__ alphap,
    const float* __restrict__ betap,
    float* __restrict__ out)         // (8, 2048, 2048)
{
    __shared__ float s_logit[ROWS * NCOL];   // 128 KB strip of logits / exps
    __shared__ float s_x[NCOL];
    __shared__ float s_y[NCOL];
    __shared__ float s_p[NCOL];              // momentum^(2*alpha) per column
    __shared__ float s_prow[ROWS];
    __shared__ float s_rowmax[ROWS];
    __shared__ float s_rowsum[ROWS];
    __shared__ float s_wmax[8 * ROWS];

    const int tid  = threadIdx.x;
    const int lane = tid & 31;
    const int wv   = tid >> 5;       // wave id 0..7
    const int half = lane >> 4;      // 0 = lanes 0-15, 1 = lanes 16-31
    const int l15  = lane & 15;

    const int blk = blockIdx.x;
    const int b   = blk >> 7;        // batch 0..7
    const int i0  = (blk & 127) << 4;

    const float alpha = alphap[0];
    const float beta  = betap[0];
    const float nb2   = -beta * beta;
    const float a2    = 2.0f * alpha;

    const float* E = emb + (size_t)b * 6 * NCOL;

    // ---- Stage columns into LDS: x, y, p = exp(2a*log(mom+eps)) ----
    #pragma unroll
    for (int k = 0; k < 8; ++k) {
        int c = (k << 8) + tid;
        float x   = E[1 * NCOL + c];
        float y   = E[2 * NCOL + c];
        float mom = E[4 * NCOL + c] + EPSF;
        s_x[c] = x;
        s_y[c] = y;
        s_p[c] = __expf(a2 * __logf(mom));
    }
    if (tid < ROWS) {
        float mom = E[4 * NCOL + i0 + tid] + EPSF;
        s_prow[tid]  = __expf(a2 * __logf(mom));
        s_rowsum[tid] = 0.0f;
    }
    __syncthreads();

    // ---- Build A fragment (16x4 f32, shared by all waves) ----
    // Rank-4 factorization of sqdist:
    //   A[i,:] = [x_i, y_i, |x_i|^2, 1],  B[:,j] = [-2x_j, -2y_j, 1, |x_j|^2]
    // A layout: lanes 0-15 hold K=0,1 in v0,v1; lanes 16-31 hold K=2,3.
    float xi = s_x[i0 + l15];
    float yi = s_y[i0 + l15];
    v2f afrag;
    afrag[0] = half ? (xi * xi + yi * yi) : xi;
    afrag[1] = half ? 1.0f : yi;

    float pi[8];
    #pragma unroll
    for (int r = 0; r < 8; ++r) pi[r] = s_prow[half * 8 + r];

    float rmax[8];
    #pragma unroll
    for (int r = 0; r < 8; ++r) rmax[r] = -INFINITY;

    // ---- Pass A: WMMA tiles -> scaled logits in LDS + running row max ----
    #pragma unroll 4
    for (int s = 0; s < 16; ++s) {
        int j0 = ((s << 3) + wv) << 4;      // this wave's column tile
        float xj = s_x[j0 + l15];
        float yj = s_y[j0 + l15];
        // B layout: lanes 0-15 hold rows K=0,1 in v0,v1; lanes 16-31 K=2,3.
        v2f bfrag;
        bfrag[0] = half ? 1.0f : (-2.0f * xj);
        bfrag[1] = half ? (xj * xj + yj * yj) : (-2.0f * yj);

        v8f c = {};
        c = __builtin_amdgcn_wmma_f32_16x16x4_f32(
                false, afrag, false, bfrag, (short)0, c, false, false);

        float pj = s_p[j0 + l15];
        #pragma unroll
        for (int r = 0; r < 8; ++r) {
            // D layout: vgpr r holds row (r + 8*half), col = j0 + (lane&15)
            float logit = nb2 * c[r] * fminf(pi[r], pj);
            rmax[r] = fmaxf(rmax[r], logit);
            s_logit[(half * 8 + r) * NCOL + j0 + l15] = logit;
        }
    }

    // ---- Row-max: reduce across the 16-lane half, then across waves ----
    #pragma unroll
    for (int r = 0; r < 8; ++r) {
        #pragma unroll
        for (int m = 1; m <= 8; m <<= 1)
            rmax[r] = fmaxf(rmax[r], __shfl_xor(rmax[r], m, 32));
    }
    if (l15 == 0) {
        #pragma unroll
        for (int r = 0; r < 8; ++r)
            s_wmax[wv * ROWS + half * 8 + r] = rmax[r];
    }
    __syncthreads();
    if (tid < ROWS) {
        float m = s_wmax[tid];
        #pragma unroll
        for (int w = 1; w < 8; ++w) m = fmaxf(m, s_wmax[w * ROWS + tid]);
        s_rowmax[tid] = m;
    }
    __syncthreads();

    // ---- Pass B: exp(logit - rowmax) in place + row sums (ds_add_f32) ----
    {
        float acc = 0.0f;
        #pragma unroll 8
        for (int k = 0; k < 128; ++k) {
            int idx = (k << 8) + tid;       // coalesced LDS banks
            int r = k >> 3;                 // 8 iterations per row
            float e = __expf(s_logit[idx] - s_rowmax[r]);
            s_logit[idx] = e;
            acc += e;
            if ((k & 7) == 7) { atomicAdd(&s_rowsum[r], acc); acc = 0.0f; }
        }
    }
    __syncthreads();

    // ---- Pass C: normalize and stream to HBM (coalesced 1KB bursts) ----
    {
        float* dst = out + (size_t)b * NCOL * NCOL + (size_t)i0 * NCOL;
        float inv = 0.0f;
        #pragma unroll 8
        for (int k = 0; k < 128; ++k) {
            int r = k >> 3;
            if ((k & 7) == 0) inv = 1.0f / s_rowsum[r];
            int idx = (k << 8) + tid;
            dst[idx] = s_logit[idx] * inv;
        }
    }
}

extern "C" void kernel_launch(void* const* d_in, const int* in_sizes, int n_in,
                              void* d_out, int out_size, void* d_ws, size_t ws_size,
                              hipStream_t stream) {
    const float* emb   = (const float*)d_in[0];
    const float* alpha = (const float*)d_in[1];
    const float* beta  = (const float*)d_in[2];
    float* out = (float*)d_out;

    dim3 grid(8 * (NCOL / ROWS));   // 1024 blocks: (batch, 16-row strip)
    dim3 block(256);                // 8 wave32
    qcd_softmax_kernel<<<grid, block, 0, stream>>>(emb, alpha, beta, out);
}